// HNPLinear_86139864088628
// MI455X (gfx1250) — compile-verified
//
#include <hip/hip_runtime.h>

// ---------------- problem constants ----------------
constexpr int kB   = 256;
constexpr int kC   = 16;
constexpr int kCO  = 16;
constexpr int kNI  = 16;
constexpr int kNO  = 10;
constexpr int kH1  = 128;
constexpr int kH2  = 128;
constexpr int kNRC = 672;   // (16 + 2*10 + 2*3) * 16

typedef __attribute__((ext_vector_type(2))) float v2f;
typedef __attribute__((ext_vector_type(8))) float v8f;

// ---------------- workspace layout (floats) ----------------
constexpr size_t ROW1_OFF  = 0;                                  // (B,16,128)
constexpr size_t COL0_OFF  = ROW1_OFF  + (size_t)kB * kC * kH1;  // (B,16,128)
constexpr size_t COL1_OFF  = COL0_OFF  + (size_t)kB * kC * kH1;  // (B,16,128)
constexpr size_t ROW2_OFF  = COL1_OFF  + (size_t)kB * kC * kH2;  // (B,16,128)
constexpr size_t RCIN_OFF  = ROW2_OFF  + (size_t)kB * kC * kH2;  // (B,672)
constexpr size_t LIN0_OFF  = RCIN_OFF  + (size_t)kB * kNRC;      // (B,256)
constexpr size_t LINB0_OFF = LIN0_OFF  + (size_t)kB * 256;       // (B,16)
constexpr size_t LIN1_OFF  = LINB0_OFF + (size_t)kB * 16;        // (B,16)
constexpr size_t LINB1_OFF = LIN1_OFF  + (size_t)kB * 16;        // (B,16)
constexpr size_t LINF_OFF  = LINB1_OFF + (size_t)kB * 16;        // (B,160)
constexpr size_t Z3_OFF    = LINF_OFF  + (size_t)kB * 160;       // (B,16,128)
constexpr size_t Z4_OFF    = Z3_OFF    + (size_t)kB * 16 * kH1;  // (B,16,128)

// ---------------- output layout (floats) ----------------
constexpr size_t ZW0_OFF = 0;
constexpr size_t ZW1_OFF = ZW0_OFF + (size_t)kB * kCO * kH1 * kNI;
constexpr size_t ZW2_OFF = ZW1_OFF + (size_t)kB * kCO * kH2 * kH1;
constexpr size_t U0_OFF  = ZW2_OFF + (size_t)kB * kCO * kNO * kH2;
constexpr size_t U1_OFF  = U0_OFF  + (size_t)kB * kCO * kH1;
constexpr size_t U2_OFF  = U1_OFF  + (size_t)kB * kCO * kH2;

// ---------------- helpers ----------------
__device__ __forceinline__ float block_reduce_128(float v, float* red) {
  const int t = threadIdx.x;
  red[t] = v;
  __syncthreads();
  for (int s = 64; s > 0; s >>= 1) {
    if (t < s) red[t] += red[t + s];
    __syncthreads();
  }
  float r = red[0];
  __syncthreads();
  return r;
}

__device__ __forceinline__ v2f ld2(const float* p) { return *(const v2f*)p; }

// ---------------- kernel 1: reductions / rc_inp ----------------
__global__ void stats_kernel(const float* __restrict__ w0, const float* __restrict__ w1,
                             const float* __restrict__ w2, const float* __restrict__ b0,
                             const float* __restrict__ b1, const float* __restrict__ b2,
                             float* __restrict__ ws) {
  __shared__ float red[128];
  const int bc = blockIdx.x;          // b*16 + c
  const int b  = bc >> 4, c = bc & 15;
  const int t  = threadIdx.x;         // 0..127
  float* rcin = ws + RCIN_OFF + (size_t)b * kNRC;

  // w1 stats
  const float* w1p = w1 + (size_t)bc * kH2 * kH1;
  float s = 0.f;
  for (int j = 0; j < kH1; ++j) s += w1p[t * kH1 + j];
  ws[COL1_OFF + (size_t)bc * kH2 + t] = s * (1.f / kH1);
  const float tot1 = block_reduce_128(s, red);
  float s2 = 0.f;
  for (int i = 0; i < kH2; ++i) s2 += w1p[i * kH1 + t];
  ws[ROW1_OFF + (size_t)bc * kH1 + t] = s2 * (1.f / kH2);

  // w0 stats
  const float* w0p = w0 + (size_t)bc * kH1 * kNI;
  float s0 = 0.f;
  for (int k = 0; k < kNI; ++k) s0 += w0p[t * kNI + k];
  ws[COL0_OFF + (size_t)bc * kH1 + t] = s0 * (1.f / kNI);
  const float tot0 = block_reduce_128(s0, red);
  if (t < kNI) {
    float sr = 0.f;
    for (int h = 0; h < kH1; ++h) sr += w0p[h * kNI + t];
    rcin[96 + c * kNI + t] = sr * (1.f / kH1);               // rm0
  }

  // w2 stats
  const float* w2p = w2 + (size_t)bc * kNO * kH2;
  float sw = 0.f;
  for (int k = 0; k < kNO; ++k) sw += w2p[k * kH2 + t];
  ws[ROW2_OFF + (size_t)bc * kH2 + t] = sw * (1.f / kNO);
  const float tot2 = block_reduce_128(sw, red);
  if (t < kNO) {
    float sc = 0.f;
    for (int h = 0; h < kH2; ++h) sc += w2p[t * kH2 + h];
    rcin[352 + c * kNO + t] = sc * (1.f / kH2);              // cmL
    rcin[512 + c * kNO + t] = b2[bc * kNO + t];              // fb
  }

  // bias means
  const float sb0 = block_reduce_128(b0[(size_t)bc * kH1 + t], red);
  const float sb1 = block_reduce_128(b1[(size_t)bc * kH2 + t], red);

  if (t == 0) {
    rcin[0 * kC + c] = tot0 * (1.f / (kH1 * kNI));
    rcin[1 * kC + c] = tot1 * (1.f / (kH2 * kH1));
    rcin[2 * kC + c] = tot2 * (1.f / (kNO * kH2));
    rcin[3 * kC + c] = sb0 * (1.f / kH1);
    rcin[4 * kC + c] = sb1 * (1.f / kH2);
    float sb2 = 0.f;
    for (int k = 0; k < kNO; ++k) sb2 += b2[bc * kNO + k];
    rcin[5 * kC + c] = sb2 * (1.f / kNO);
  }
}

// ---------------- kernel 2: rc_inp linear projections ----------------
__global__ void rclin_kernel(const float* __restrict__ W_w0_rc,    const float* __restrict__ bb_w0_rc,
                             const float* __restrict__ W_bias0_rc, const float* __restrict__ bb_bias0_rc,
                             const float* __restrict__ W_l1_rc,    const float* __restrict__ bb_l1_rc,
                             const float* __restrict__ W_bias1_rc, const float* __restrict__ bb_bias1_rc,
                             const float* __restrict__ W_fin_rc,   const float* __restrict__ bb_fin_rc,
                             const float* __restrict__ W_bfin_rc,  const float* __restrict__ bb_bfin_rc,
                             float* __restrict__ ws, float* __restrict__ out) {
  __shared__ float rin[kNRC];
  const int b = blockIdx.x;
  for (int i = threadIdx.x; i < kNRC; i += blockDim.x)
    rin[i] = ws[RCIN_OFF + (size_t)b * kNRC + i];
  __syncthreads();

  for (int idx = threadIdx.x; idx < 624; idx += blockDim.x) {
    const float* Wr; float bias; float* dst;
    if      (idx < 256) { int o = idx;       Wr = W_w0_rc    + (size_t)o * kNRC; bias = bb_w0_rc[o];    dst = ws + LIN0_OFF  + (size_t)b * 256 + o; }
    else if (idx < 272) { int o = idx - 256; Wr = W_bias0_rc + (size_t)o * kNRC; bias = bb_bias0_rc[o]; dst = ws + LINB0_OFF + (size_t)b * 16  + o; }
    else if (idx < 288) { int o = idx - 272; Wr = W_l1_rc    + (size_t)o * kNRC; bias = bb_l1_rc[o];    dst = ws + LIN1_OFF  + (size_t)b * 16  + o; }
    else if (idx < 304) { int o = idx - 288; Wr = W_bias1_rc + (size_t)o * kNRC; bias = bb_bias1_rc[o]; dst = ws + LINB1_OFF + (size_t)b * 16  + o; }
    else if (idx < 464) { int o = idx - 304; Wr = W_fin_rc   + (size_t)o * kNRC; bias = bb_fin_rc[o];   dst = ws + LINF_OFF  + (size_t)b * 160 + o; }
    else                { int o = idx - 464; Wr = W_bfin_rc  + (size_t)o * kNRC; bias = bb_bfin_rc[o];  dst = out + U2_OFF   + (size_t)b * 160 + o; }
    float acc = bias;
    for (int k = 0; k < kNRC; ++k) acc += Wr[k] * rin[k];
    *dst = acc;
  }
}

// ---------------- kernel 3: thin convs (z3, z4, u0, u1), segment-split ----------------
__global__ void smallconv_kernel(const float* __restrict__ w0, const float* __restrict__ w2,
                                 const float* __restrict__ b0, const float* __restrict__ b1,
                                 const float* __restrict__ W_bias0, const float* __restrict__ bb_bias0,
                                 const float* __restrict__ W_l1_r,  const float* __restrict__ bb_l1_r,
                                 const float* __restrict__ W_l1_c,  const float* __restrict__ bb_l1_c,
                                 const float* __restrict__ W_bias1, const float* __restrict__ bb_bias1,
                                 float* __restrict__ ws, float* __restrict__ out) {
  const int b = blockIdx.x;
  const int n = threadIdx.x;   // 0..127

  const float* r1b = ws + ROW1_OFF + (size_t)b * 2048 + n;   // row1[b,c,n], stride 128
  const float* c0b = ws + COL0_OFF + (size_t)b * 2048 + n;
  const float* c1b = ws + COL1_OFF + (size_t)b * 2048 + n;
  const float* r2b = ws + ROW2_OFF + (size_t)b * 2048 + n;
  const float* b0b = b0 + (size_t)b * 2048 + n;
  const float* b1b = b1 + (size_t)b * 2048 + n;
  const float* w0b = w0 + (size_t)b * 32768 + (size_t)n * 16; // w0_t row (c*16+j) -> w0b[c*2048+j]
  const float* w2b = w2 + (size_t)b * 20480 + n;              // w2_f row r -> w2b[r*128]

  // z3 = conv1(row_bd = [row1, col0, b0, w0_t], W_l1_r)
  {
    float acc[16];
    #pragma unroll
    for (int o = 0; o < 16; ++o) acc[o] = 0.f;
    for (int r = 0; r < 16; ++r) {
      const float v = r1b[r * 128];
      #pragma unroll
      for (int o = 0; o < 16; ++o) acc[o] += W_l1_r[o * 304 + r] * v;
    }
    for (int r = 0; r < 16; ++r) {
      const float v = c0b[r * 128];
      #pragma unroll
      for (int o = 0; o < 16; ++o) acc[o] += W_l1_r[o * 304 + 16 + r] * v;
    }
    for (int r = 0; r < 16; ++r) {
      const float v = b0b[r * 128];
      #pragma unroll
      for (int o = 0; o < 16; ++o) acc[o] += W_l1_r[o * 304 + 32 + r] * v;
    }
    for (int c = 0; c < 16; ++c)
      for (int j = 0; j < 16; ++j) {
        const float v = w0b[c * 2048 + j];
        #pragma unroll
        for (int o = 0; o < 16; ++o) acc[o] += W_l1_r[o * 304 + 48 + c * 16 + j] * v;
      }
    #pragma unroll
    for (int o = 0; o < 16; ++o)
      ws[Z3_OFF + ((size_t)b * 16 + o) * kH1 + n] = acc[o] + bb_l1_r[o];
  }

  // z4 + u1 over col_bd = [col1, b1, row2, w2_f]
  {
    float a4[16], au[16];
    #pragma unroll
    for (int o = 0; o < 16; ++o) { a4[o] = 0.f; au[o] = 0.f; }
    for (int r = 0; r < 16; ++r) {
      const float v = c1b[r * 128];
      #pragma unroll
      for (int o = 0; o < 16; ++o) {
        a4[o] += W_l1_c[o * 208 + r] * v;
        au[o] += W_bias1[o * 208 + r] * v;
      }
    }
    for (int r = 0; r < 16; ++r) {
      const float v = b1b[r * 128];
      #pragma unroll
      for (int o = 0; o < 16; ++o) {
        a4[o] += W_l1_c[o * 208 + 16 + r] * v;
        au[o] += W_bias1[o * 208 + 16 + r] * v;
      }
    }
    for (int r = 0; r < 16; ++r) {
      const float v = r2b[r * 128];
      #pragma unroll
      for (int o = 0; o < 16; ++o) {
        a4[o] += W_l1_c[o * 208 + 32 + r] * v;
        au[o] += W_bias1[o * 208 + 32 + r] * v;
      }
    }
    for (int r = 0; r < 160; ++r) {
      const float v = w2b[(size_t)r * 128];
      #pragma unroll
      for (int o = 0; o < 16; ++o) {
        a4[o] += W_l1_c[o * 208 + 48 + r] * v;
        au[o] += W_bias1[o * 208 + 48 + r] * v;
      }
    }
    #pragma unroll
    for (int o = 0; o < 16; ++o) {
      ws[Z4_OFF + ((size_t)b * 16 + o) * kH2 + n] = a4[o] + bb_l1_c[o];
      out[U1_OFF + ((size_t)b * 16 + o) * kH2 + n] =
          au[o] + bb_bias1[o] + ws[LINB1_OFF + (size_t)b * 16 + o];
    }
  }

  // u0 over rpt = [w0_t, row1, b0]
  {
    float acc[16];
    #pragma unroll
    for (int o = 0; o < 16; ++o) acc[o] = 0.f;
    for (int c = 0; c < 16; ++c)
      for (int j = 0; j < 16; ++j) {
        const float v = w0b[c * 2048 + j];
        #pragma unroll
        for (int o = 0; o < 16; ++o) acc[o] += W_bias0[o * 288 + c * 16 + j] * v;
      }
    for (int r = 0; r < 16; ++r) {
      const float v = r1b[r * 128];
      #pragma unroll
      for (int o = 0; o < 16; ++o) acc[o] += W_bias0[o * 288 + 256 + r] * v;
    }
    for (int r = 0; r < 16; ++r) {
      const float v = b0b[r * 128];
      #pragma unroll
      for (int o = 0; o < 16; ++o) acc[o] += W_bias0[o * 288 + 272 + r] * v;
    }
    #pragma unroll
    for (int o = 0; o < 16; ++o)
      out[U0_OFF + ((size_t)b * 16 + o) * kH1 + n] =
          acc[o] + bb_bias0[o] + ws[LINB0_OFF + (size_t)b * 16 + o];
  }
}

// ---------------- kernel 4: layer0 GEMM (256x288)x(288x128), segment-split ----------------
__global__ void layer0_wmma(const float* __restrict__ w0, const float* __restrict__ b0,
                            const float* __restrict__ W_w0_rpt, const float* __restrict__ bb_w0_rpt,
                            const float* __restrict__ ws, float* __restrict__ out) {
  const int wave = threadIdx.x >> 5;
  const int lane = threadIdx.x & 31;
  const int hl = lane >> 4, ln = lane & 15;
  const int gtile  = blockIdx.x * 8 + wave;   // kB*128 tiles
  const int b      = gtile >> 7;
  const int t      = gtile & 127;
  const int m_base = (t >> 3) * 16;
  const int n      = (t & 7) * 16 + ln;
  const int m      = m_base + ln;

  const float* Arow = W_w0_rpt + (size_t)m * 288;
  const float* w0b  = w0 + (size_t)b * 32768 + (size_t)n * 16;   // + c*2048 + j
  const float* r1b  = ws + ROW1_OFF + (size_t)b * 2048 + n;      // + c*128
  const float* b0b  = b0 + (size_t)b * 2048 + n;                 // + c*128
  const int h2 = 2 * hl;

  v8f acc = {0.f, 0.f, 0.f, 0.f, 0.f, 0.f, 0.f, 0.f};

  // segment 1: k in [0,256) -> w0_t rows (contiguous float2 per row-block)
  for (int c = 0; c < 16; ++c) {
    const float* wsrc = w0b + c * 2048;
    #pragma unroll
    for (int q = 0; q < 4; ++q) {
      const int j = q * 4 + h2;                 // 0..14, even
      v2f av = ld2(Arow + c * 16 + j);
      v2f bv = ld2(wsrc + j);
      acc = __builtin_amdgcn_wmma_f32_16x16x4_f32(false, av, false, bv,
                                                  (short)0, acc, false, false);
    }
  }
  // segment 2: k in [256,272) -> row1
  #pragma unroll
  for (int q = 0; q < 4; ++q) {
    const int cc = q * 4 + h2;
    v2f av = ld2(Arow + 256 + cc);
    v2f bv; bv.x = r1b[cc * 128]; bv.y = r1b[(cc + 1) * 128];
    acc = __builtin_amdgcn_wmma_f32_16x16x4_f32(false, av, false, bv,
                                                (short)0, acc, false, false);
  }
  // segment 3: k in [272,288) -> b0
  #pragma unroll
  for (int q = 0; q < 4; ++q) {
    const int cc = q * 4 + h2;
    v2f av = ld2(Arow + 272 + cc);
    v2f bv; bv.x = b0b[cc * 128]; bv.y = b0b[(cc + 1) * 128];
    acc = __builtin_amdgcn_wmma_f32_16x16x4_f32(false, av, false, bv,
                                                (short)0, acc, false, false);
  }

  #pragma unroll
  for (int r = 0; r < 8; ++r) {
    const int mm = m_base + r + hl * 8;
    const float v = acc[r] + bb_w0_rpt[mm] + ws[LIN0_OFF + (size_t)b * 256 + mm];
    // zw0[b, mm>>4, n, mm&15]  (swapaxes fused into the store)
    out[ZW0_OFF + (((size_t)b * kCO + (mm >> 4)) * kH1 + n) * kNI + (mm & 15)] = v;
  }
}

// ---------------- kernel 5: layer1 GEMM (16x16)x(16x16384) per batch ----------------
__global__ void layer1_wmma(const float* __restrict__ w1, const float* __restrict__ W_l1,
                            const float* __restrict__ bb_l1, const float* __restrict__ ws,
                            float* __restrict__ out) {
  const int wave = threadIdx.x >> 5;
  const int lane = threadIdx.x & 31;
  const int hl = lane >> 4, ln = lane & 15;
  const int gtile = blockIdx.x * 8 + wave;    // kB*1024 tiles
  const int b     = gtile >> 10;
  const int p     = (gtile & 1023) * 16 + ln; // p = i*128 + j
  const float* w1b = w1 + (size_t)b * 16 * 16384 + p;

  v8f acc = {0.f, 0.f, 0.f, 0.f, 0.f, 0.f, 0.f, 0.f};
  #pragma unroll
  for (int k0 = 0; k0 < 16; k0 += 4) {
    const int ka = k0 + 2 * hl;
    v2f av = ld2(W_l1 + ln * 16 + ka);
    v2f bv;
    bv.x = w1b[(size_t)ka * 16384];
    bv.y = w1b[(size_t)(ka + 1) * 16384];
    acc = __builtin_amdgcn_wmma_f32_16x16x4_f32(false, av, false, bv,
                                                (short)0, acc, false, false);
  }
  const int i = p >> 7, j = p & 127;
  #pragma unroll
  for (int r = 0; r < 8; ++r) {
    const int o = r + hl * 8;
    const float v = acc[r] + bb_l1[o]
                  + ws[LIN1_OFF + (size_t)b * 16 + o]
                  + ws[Z3_OFF + ((size_t)b * 16 + o) * kH1 + j]
                  + ws[Z4_OFF + ((size_t)b * 16 + o) * kH2 + i];
    out[ZW1_OFF + ((size_t)b * 16 + o) * 16384 + p] = v;
  }
}

// ---------------- kernel 6: layer2 GEMM (160x176)x(176x128), segment-split ----------------
__global__ void layer2_wmma(const float* __restrict__ w2, const float* __restrict__ W_fin_cpt,
                            const float* __restrict__ bb_fin_cpt, const float* __restrict__ ws,
                            float* __restrict__ out) {
  const int wave = threadIdx.x >> 5;
  const int lane = threadIdx.x & 31;
  const int hl = lane >> 4, ln = lane & 15;
  const int gtile  = blockIdx.x * 8 + wave;   // kB*80 tiles
  const int b      = gtile / 80;
  const int t      = gtile % 80;
  const int m_base = (t >> 3) * 16;
  const int n      = (t & 7) * 16 + ln;
  const int m      = m_base + ln;

  const float* Arow = W_fin_cpt + (size_t)m * 176;
  const float* w2b  = w2 + (size_t)b * 20480 + n;            // w2_f row r -> w2b[r*128]
  const float* c1b  = ws + COL1_OFF + (size_t)b * 2048 + n;  // col1 row c -> c1b[c*128]
  const int h2 = 2 * hl;

  v8f acc = {0.f, 0.f, 0.f, 0.f, 0.f, 0.f, 0.f, 0.f};

  // segment 1: k in [0,160) -> w2_f
  for (int k0 = 0; k0 < 160; k0 += 4) {
    const int r = k0 + h2;
    v2f av = ld2(Arow + r);
    v2f bv; bv.x = w2b[(size_t)r * 128]; bv.y = w2b[(size_t)(r + 1) * 128];
    acc = __builtin_amdgcn_wmma_f32_16x16x4_f32(false, av, false, bv,
                                                (short)0, acc, false, false);
  }
  // segment 2: k in [160,176) -> col1
  #pragma unroll
  for (int q = 0; q < 4; ++q) {
    const int cc = q * 4 + h2;
    v2f av = ld2(Arow + 160 + cc);
    v2f bv; bv.x = c1b[cc * 128]; bv.y = c1b[(cc + 1) * 128];
    acc = __builtin_amdgcn_wmma_f32_16x16x4_f32(false, av, false, bv,
                                                (short)0, acc, false, false);
  }

  #pragma unroll
  for (int r = 0; r < 8; ++r) {
    const int o = m_base + r + hl * 8;     // o = co*10 + no
    const float v = acc[r] + bb_fin_cpt[o] + ws[LINF_OFF + (size_t)b * 160 + o];
    out[ZW2_OFF + ((size_t)b * 160 + o) * kH2 + n] = v;
  }
}

// ---------------- launcher ----------------
extern "C" void kernel_launch(void* const* d_in, const int* in_sizes, int n_in,
                              void* d_out, int out_size, void* d_ws, size_t ws_size,
                              hipStream_t stream) {
  (void)in_sizes; (void)n_in; (void)out_size; (void)ws_size;
  const float* w0          = (const float*)d_in[0];
  const float* w1          = (const float*)d_in[1];
  const float* w2          = (const float*)d_in[2];
  const float* b0          = (const float*)d_in[3];
  const float* b1          = (const float*)d_in[4];
  const float* b2          = (const float*)d_in[5];
  const float* W_w0_rpt    = (const float*)d_in[6];
  const float* bb_w0_rpt   = (const float*)d_in[7];
  const float* W_w0_rc     = (const float*)d_in[8];
  const float* bb_w0_rc    = (const float*)d_in[9];
  const float* W_bias0     = (const float*)d_in[10];
  const float* bb_bias0    = (const float*)d_in[11];
  const float* W_bias0_rc  = (const float*)d_in[12];
  const float* bb_bias0_rc = (const float*)d_in[13];
  const float* W_l1        = (const float*)d_in[14];
  const float* bb_l1       = (const float*)d_in[15];
  const float* W_l1_rc     = (const float*)d_in[16];
  const float* bb_l1_rc    = (const float*)d_in[17];
  const float* W_l1_r      = (const float*)d_in[18];
  const float* bb_l1_r     = (const float*)d_in[19];
  const float* W_l1_c      = (const float*)d_in[20];
  const float* bb_l1_c     = (const float*)d_in[21];
  const float* W_bias1     = (const float*)d_in[22];
  const float* bb_bias1    = (const float*)d_in[23];
  const float* W_bias1_rc  = (const float*)d_in[24];
  const float* bb_bias1_rc = (const float*)d_in[25];
  const float* W_fin_cpt   = (const float*)d_in[26];
  const float* bb_fin_cpt  = (const float*)d_in[27];
  const float* W_fin_rc    = (const float*)d_in[28];
  const float* bb_fin_rc   = (const float*)d_in[29];
  const float* W_bfin_rc   = (const float*)d_in[30];
  const float* bb_bfin_rc  = (const float*)d_in[31];
  float* out = (float*)d_out;
  float* ws  = (float*)d_ws;

  stats_kernel<<<kB * kC, 128, 0, stream>>>(w0, w1, w2, b0, b1, b2, ws);
  rclin_kernel<<<kB, 256, 0, stream>>>(W_w0_rc, bb_w0_rc, W_bias0_rc, bb_bias0_rc,
                                       W_l1_rc, bb_l1_rc, W_bias1_rc, bb_bias1_rc,
                                       W_fin_rc, bb_fin_rc, W_bfin_rc, bb_bfin_rc,
                                       ws, out);
  smallconv_kernel<<<kB, 128, 0, stream>>>(w0, w2, b0, b1, W_bias0, bb_bias0,
                                           W_l1_r, bb_l1_r, W_l1_c, bb_l1_c,
                                           W_bias1, bb_bias1, ws, out);
  layer0_wmma<<<kB * 16, 256, 0, stream>>>(w0, b0, W_w0_rpt, bb_w0_rpt, ws, out);
  layer1_wmma<<<kB * 128, 256, 0, stream>>>(w1, W_l1, bb_l1, ws, out);
  layer2_wmma<<<kB * 10, 256, 0, stream>>>(w2, W_fin_cpt, bb_fin_cpt, ws, out);
}